// Decoder_14139032339002
// MI455X (gfx1250) — compile-verified
//
#include <hip/hip_runtime.h>
#include <hip/hip_bf16.h>

typedef float v2f __attribute__((ext_vector_type(2)));
typedef float v8f __attribute__((ext_vector_type(8)));

#define B_IMG 32
#define HW 64
#define F 64
#define NCH 256          // 4*F gate channels
#define KDIM 576         // 9*F im2col depth
#define TILE_M 32        // pixels per block (half a row)
#define LDS_W 34         // 32 + 2 halo columns

__device__ __forceinline__ float hsig(float x) {
    return fminf(fmaxf(0.2f * x + 0.5f, 0.0f), 1.0f);
}

// Stage a 3x34x64 fp32 halo tile of src (layout [B,64,64,64]) into LDS, zero-padded.
__device__ __forceinline__ void stage_tile(const float* __restrict__ src, float* hl,
                                           int bb, int y, int x0, int tid) {
    // 3*34*64 floats = 1632 float4s
    for (int i = tid; i < 3 * LDS_W * 16; i += 256) {
        int r = i / (LDS_W * 16);
        int rem = i - r * (LDS_W * 16);
        int xi = rem >> 4;
        int c4 = rem & 15;
        int yy = y + r - 1;
        int xx = x0 + xi - 1;
        float4 v = make_float4(0.f, 0.f, 0.f, 0.f);
        if ((unsigned)yy < 64u && (unsigned)xx < 64u) {
            v = *(const float4*)(src + ((((bb * HW) + yy) * HW + xx) * F + c4 * 4));
        }
        *(float4*)(hl + ((r * LDS_W + xi) * F + c4 * 4)) = v;
    }
}

// One implicit-GEMM pass: acc[4] += A(32x576 from LDS im2col) x W(576x256), this wave's tiles.
__device__ __forceinline__ void gemm_pass(const float* __restrict__ W, const float* hl,
                                          v8f acc[4], int mtile, int jset, int lane) {
    const int m      = lane & 15;
    const int khalf  = lane >> 4;
    const int xl     = mtile * 16 + m;          // A-row -> local pixel column
    const int ncol   = jset * 64 + (lane & 15); // base N column for j=0

    const int kyA[9] = {0, 0, 0, 1, 1, 1, 2, 2, 2};
    const int kxA[9] = {0, 1, 2, 0, 1, 2, 0, 1, 2};

    for (int tap = 0; tap < 9; ++tap) {
        const float* arow  = hl + ((kyA[tap] * LDS_W) + (xl + kxA[tap])) * F;
        const float* wbase = W + (tap * F) * NCH;
        if (tap < 8) __builtin_prefetch(W + ((tap + 1) * F) * NCH, 0, 1);
        for (int c4 = 0; c4 < 16; ++c4) {
            const int cb = c4 * 4;
            // A fragment: lanes 0-15 hold K={cb,cb+1}, lanes 16-31 hold K={cb+2,cb+3}
            v2f a = *(const v2f*)(arow + cb + 2 * khalf);
#pragma unroll
            for (int j = 0; j < 4; ++j) {
                const float* wp = wbase + (cb + 2 * khalf) * NCH + (ncol + j * 16);
                v2f bf;
                bf.x = wp[0];
                bf.y = wp[NCH];
                acc[j] = __builtin_amdgcn_wmma_f32_16x16x4_f32(
                    false, a, false, bf, (short)0, acc[j], false, false);
            }
        }
    }
}

__device__ __forceinline__ void store_z(float* zbuf, v8f acc[4], int mtile, int jset, int lane) {
    const int half   = lane >> 4;
    const int lane16 = lane & 15;
#pragma unroll
    for (int j = 0; j < 4; ++j) {
        const int n = jset * 64 + j * 16 + lane16;
#pragma unroll
        for (int v = 0; v < 8; ++v) {
            const int p = mtile * 16 + v + 8 * half;
            zbuf[p * NCH + n] = acc[j][v];
        }
    }
}

// ---------------- Layer 0: z = conv3x3(h0, rk0) + conv3x3(input Cin=1, k0) + b0 ----------------
__global__ void __launch_bounds__(256)
convlstm_layer0(const float* __restrict__ in_, const float* __restrict__ h0,
                const float* __restrict__ c0, const float* __restrict__ k0,
                const float* __restrict__ rk0, const float* __restrict__ b0,
                float* __restrict__ hOut, float* __restrict__ cOut) {
    __shared__ __align__(16) float smem[8192 + 112]; // zbuf(8192) aliases h_lds(6528); xin after
    float* hl   = smem;
    float* zbuf = smem;
    float* xin  = smem + 8192;

    const int tid   = threadIdx.x;
    const int lane  = tid & 31;
    const int wave  = tid >> 5;
    const int mtile = wave >> 2;
    const int jset  = wave & 3;

    const int bid = blockIdx.x;
    const int x0  = (bid & 1) * TILE_M;
    const int y   = (bid >> 1) & 63;
    const int bb  = bid >> 7;

    stage_tile(h0, hl, bb, y, x0, tid);
    // Cin=1 input halo tile (3x34)
    for (int i = tid; i < 3 * LDS_W; i += 256) {
        int r = i / LDS_W, xi = i - r * LDS_W;
        int yy = y + r - 1, xx = x0 + xi - 1;
        xin[i] = ((unsigned)yy < 64u && (unsigned)xx < 64u)
                     ? in_[(bb * HW + yy) * HW + xx] : 0.f;
    }
    __syncthreads();

    v8f acc[4];
#pragma unroll
    for (int j = 0; j < 4; ++j)
#pragma unroll
        for (int v = 0; v < 8; ++v) acc[j][v] = 0.f;

    gemm_pass(rk0, hl, acc, mtile, jset, lane);

    // Cin=1 input conv contribution via VALU (K=9, too small for WMMA)
    {
        const int kyA[9] = {0, 0, 0, 1, 1, 1, 2, 2, 2};
        const int kxA[9] = {0, 1, 2, 0, 1, 2, 0, 1, 2};
        const int half = lane >> 4, lane16 = lane & 15;
        float wj[4][9];
#pragma unroll
        for (int j = 0; j < 4; ++j)
#pragma unroll
            for (int t = 0; t < 9; ++t)
                wj[j][t] = k0[t * NCH + jset * 64 + j * 16 + lane16];
#pragma unroll
        for (int v = 0; v < 8; ++v) {
            const int xl = mtile * 16 + v + 8 * half;
            float xv[9];
#pragma unroll
            for (int t = 0; t < 9; ++t) xv[t] = xin[kyA[t] * LDS_W + xl + kxA[t]];
#pragma unroll
            for (int j = 0; j < 4; ++j)
#pragma unroll
                for (int t = 0; t < 9; ++t) acc[j][v] += xv[t] * wj[j][t];
        }
    }
    __syncthreads();             // all LDS A-tile reads done before aliasing as zbuf
    store_z(zbuf, acc, mtile, jset, lane);
    __syncthreads();

    // Gates: i,f,o = hard_sigmoid; c' = f*c + i*tanh(zc); h' = o*tanh(c')
#pragma unroll
    for (int r = 0; r < 8; ++r) {
        const int flat = r * 256 + tid;
        const int p = flat >> 6, n = flat & 63;
        const float zi = zbuf[p * NCH + n]        + b0[n];
        const float zf = zbuf[p * NCH + 64 + n]   + b0[64 + n];
        const float zc = zbuf[p * NCH + 128 + n]  + b0[128 + n];
        const float zo = zbuf[p * NCH + 192 + n]  + b0[192 + n];
        const int gi = (((bb * HW) + y) * HW + x0 + p) * F + n;
        const float cn = hsig(zf) * c0[gi] + hsig(zi) * tanhf(zc);
        const float hn = hsig(zo) * tanhf(cn);
        cOut[gi] = cn;
        hOut[gi] = hn;
    }
}

// ------- Layer 1: z = conv3x3(h0n, k1) + conv3x3(h1, rk1) + b1 ; fused 1x1 head -------
__global__ void __launch_bounds__(256)
convlstm_layer1(const float* __restrict__ x_in, const float* __restrict__ h1,
                const float* __restrict__ c1, const float* __restrict__ k1,
                const float* __restrict__ rk1, const float* __restrict__ b1,
                const float* __restrict__ cw, const float* __restrict__ cb,
                float* __restrict__ hOut, float* __restrict__ cOut,
                float* __restrict__ frames) {
    __shared__ __align__(16) float smem[8192 + 2048]; // zbuf(8192, aliases h_lds) + hbuf(2048)
    float* hl   = smem;
    float* zbuf = smem;
    float* hbuf = smem + 8192;

    const int tid   = threadIdx.x;
    const int lane  = tid & 31;
    const int wave  = tid >> 5;
    const int mtile = wave >> 2;
    const int jset  = wave & 3;

    const int bid = blockIdx.x;
    const int x0  = (bid & 1) * TILE_M;
    const int y   = (bid >> 1) & 63;
    const int bb  = bid >> 7;

    v8f acc[4];
#pragma unroll
    for (int j = 0; j < 4; ++j)
#pragma unroll
        for (int v = 0; v < 8; ++v) acc[j][v] = 0.f;

    // Pass 0: input conv over layer-0 hidden state
    stage_tile(x_in, hl, bb, y, x0, tid);
    __syncthreads();
    gemm_pass(k1, hl, acc, mtile, jset, lane);
    __syncthreads();

    // Pass 1: recurrent conv over h1
    stage_tile(h1, hl, bb, y, x0, tid);
    __syncthreads();
    gemm_pass(rk1, hl, acc, mtile, jset, lane);
    __syncthreads();

    store_z(zbuf, acc, mtile, jset, lane);
    __syncthreads();

#pragma unroll
    for (int r = 0; r < 8; ++r) {
        const int flat = r * 256 + tid;
        const int p = flat >> 6, n = flat & 63;
        const float zi = zbuf[p * NCH + n]        + b1[n];
        const float zf = zbuf[p * NCH + 64 + n]   + b1[64 + n];
        const float zc = zbuf[p * NCH + 128 + n]  + b1[128 + n];
        const float zo = zbuf[p * NCH + 192 + n]  + b1[192 + n];
        const int gi = (((bb * HW) + y) * HW + x0 + p) * F + n;
        const float cn = hsig(zf) * c1[gi] + hsig(zi) * tanhf(zc);
        const float hn = hsig(zo) * tanhf(cn);
        cOut[gi] = cn;
        hOut[gi] = hn;
        hbuf[p * F + n] = hn;
    }
    __syncthreads();

    // Fused 1x1 conv head: frames[p] = cb + sum_c h[p,c] * cw[c]
    if (tid < TILE_M) {
        const int p = tid;
        float s = cb[0];
#pragma unroll
        for (int c = 0; c < F; ++c) s += hbuf[p * F + c] * cw[c];
        frames[((bb * HW) + y) * HW + x0 + p] = s;
    }
}

extern "C" void kernel_launch(void* const* d_in, const int* in_sizes, int n_in,
                              void* d_out, int out_size, void* d_ws, size_t ws_size,
                              hipStream_t stream) {
    (void)in_sizes; (void)n_in; (void)out_size; (void)d_ws; (void)ws_size;
    const float* input_ = (const float*)d_in[0];
    const float* h0     = (const float*)d_in[1];
    const float* c0     = (const float*)d_in[2];
    const float* h1     = (const float*)d_in[3];
    const float* c1     = (const float*)d_in[4];
    const float* k0     = (const float*)d_in[5];
    const float* rk0    = (const float*)d_in[6];
    const float* b0     = (const float*)d_in[7];
    const float* k1     = (const float*)d_in[8];
    const float* rk1    = (const float*)d_in[9];
    const float* b1     = (const float*)d_in[10];
    const float* cw     = (const float*)d_in[11];
    const float* cb     = (const float*)d_in[12];

    float* out    = (float*)d_out;
    float* frames = out;                                   // [32,64,64,1]
    float* h0n    = out + (size_t)B_IMG * HW * HW;         // [32,64,64,64]
    float* c0n    = h0n + (size_t)B_IMG * HW * HW * F;
    float* h1n    = c0n + (size_t)B_IMG * HW * HW * F;
    float* c1n    = h1n + (size_t)B_IMG * HW * HW * F;

    const int blocks = B_IMG * HW * 2; // (batch, row, half-row) tiles
    convlstm_layer0<<<blocks, 256, 0, stream>>>(input_, h0, c0, k0, rk0, b0, h0n, c0n);
    convlstm_layer1<<<blocks, 256, 0, stream>>>(h0n, h1, c1, k1, rk1, b1, cw, cb,
                                                h1n, c1n, frames);
}